// QuantumLayer_3959959847573
// MI455X (gfx1250) — compile-verified
//
#include <hip/hip_runtime.h>
#include <stdint.h>

// ---------------------------------------------------------------------------
// QuantumLayer on MI455X (gfx1250, wave32).
//
// Reference collapses to:
//   state = [E, O, 0, ..., 0] per row  (E,O from a 78-step 2x2 complex
//   recurrence over rx/ry/rz angles, with the reference's aliasing bug),
//   plus complex Gaussian noise (JAX threefry2x32, key 42) * 0.01,
//   then per-row L2 normalize, output = abs(state).
//
// Fused single kernel: 1 block (256 thr, 8 wave32) per batch row.
//   - each thread: 32 elements -> 64 threefry2x32 calls -> erfinv -> |z|
//     magnitudes kept in registers, partial sum of squares in a VGPR
//   - block norm reduction: v_wmma_f32_16x16x4_f32 (A=ones) folds lane pairs
//     (L, L+16), then shfl_xor tree, then tiny LDS cross-wave combine
//   - coalesced float4 stores of mag * rsqrt(norm2)
// Write-bandwidth bound: 8 MB out @ 23.3 TB/s ~ 0.35 us; all else trivial.
// ---------------------------------------------------------------------------

typedef __attribute__((ext_vector_type(2))) float v2f;
typedef __attribute__((ext_vector_type(8))) float v8f;

#define N_ANG   26        // 13 qubits * 2 rotations, flattened row-major
#define DIM     8192
#define THREADS 256
#define DECO    0.01f

struct U2 { uint32_t a, b; };

__device__ __forceinline__ uint32_t rotl32(uint32_t v, uint32_t n) {
  return (v << n) | (v >> (32u - n));
}

// JAX / Random123 Threefry-2x32, 20 rounds, key schedule [k0, k1, k0^k1^C].
__device__ __forceinline__ U2 threefry2x32(uint32_t k0, uint32_t k1,
                                           uint32_t x0, uint32_t x1) {
  const uint32_t ks2 = k0 ^ k1 ^ 0x1BD11BDAu;
  x0 += k0; x1 += k1;
#define TFR(r) { x0 += x1; x1 = rotl32(x1, r); x1 ^= x0; }
  TFR(13) TFR(15) TFR(26) TFR(6)
  x0 += k1;  x1 += ks2 + 1u;
  TFR(17) TFR(29) TFR(16) TFR(24)
  x0 += ks2; x1 += k0 + 2u;
  TFR(13) TFR(15) TFR(26) TFR(6)
  x0 += k0;  x1 += k1 + 3u;
  TFR(17) TFR(29) TFR(16) TFR(24)
  x0 += k1;  x1 += ks2 + 4u;
  TFR(13) TFR(15) TFR(26) TFR(6)
  x0 += ks2; x1 += k0 + 5u;
#undef TFR
  U2 r; r.a = x0; r.b = x1; return r;
}

// JAX uniform(key,...,lo=nextafter(-1,0),hi=1) from raw 32-bit draws.
__device__ __forceinline__ float jax_uniform_pm1(uint32_t bits) {
  const float lo = -0.99999994f;                    // nextafter(-1f, 0)
  float f01 = __uint_as_float((bits >> 9) | 0x3F800000u) - 1.0f;  // [0,1)
  return fmaxf(lo, fmaf(f01, 1.0f - lo, lo));
}

// Giles (2010) single-precision erfinv.
__device__ __forceinline__ float erfinv_f(float x) {
  float w = -__logf((1.0f - x) * (1.0f + x));
  float p;
  if (w < 5.0f) {
    w -= 2.5f;
    p = 2.81022636e-08f;
    p = fmaf(p, w, 3.43273939e-07f);
    p = fmaf(p, w, -3.5233877e-06f);
    p = fmaf(p, w, -4.39150654e-06f);
    p = fmaf(p, w, 0.00021858087f);
    p = fmaf(p, w, -0.00125372503f);
    p = fmaf(p, w, -0.00417768164f);
    p = fmaf(p, w, 0.246640727f);
    p = fmaf(p, w, 1.50140941f);
  } else {
    w = __fsqrt_rn(w) - 3.0f;
    p = -0.000200214257f;
    p = fmaf(p, w, 0.000100950558f);
    p = fmaf(p, w, 0.00134934322f);
    p = fmaf(p, w, -0.00367342844f);
    p = fmaf(p, w, 0.00573950773f);
    p = fmaf(p, w, -0.0076224613f);
    p = fmaf(p, w, 0.00943887047f);
    p = fmaf(p, w, 1.00167406f);
    p = fmaf(p, w, 2.83297682f);
  }
  return p * x;
}

__global__ __launch_bounds__(THREADS)
void quantum_layer_kernel(const float* __restrict__ rx,
                          const float* __restrict__ ry,
                          const float* __restrict__ rz,
                          float* __restrict__ out,
                          uint32_t half_n) {
  const int b = blockIdx.x;
  const int t = threadIdx.x;

  __builtin_prefetch(rx, 0, 3);   // global_prefetch_b8
  __builtin_prefetch(ry, 0, 3);
  __builtin_prefetch(rz, 0, 3);

  // ---- 1. Derive JAX split keys for key(42): counts iota(4) -> (2,2) keys.
  const U2 s02 = threefry2x32(0u, 42u, 0u, 2u);
  const U2 s13 = threefry2x32(0u, 42u, 1u, 3u);
  const uint32_t kre0 = s02.a, kre1 = s13.a;   // key_re
  const uint32_t kim0 = s02.b, kim1 = s13.b;   // key_im

  // ---- 2. (E,O) amplitude pair: 78 modified 2x2 complex gates (uniform).
  float ex = 1.0f, ey = 0.0f, ox = 0.0f, oy = 0.0f;
  for (int i = 0; i < N_ANG; ++i) {            // RX: [[c,-is],[-is,c]]
    float sa, ca; __sincosf(0.5f * rx[i], &sa, &ca);
    float enx = fmaf(ca, ex,  sa * oy), eny = fmaf(ca, ey, -sa * ox);
    float onx = fmaf(sa, eny,  ca * ox), ony = fmaf(-sa, enx, ca * oy);
    ex = enx; ey = eny; ox = onx; oy = ony;    // aliasing bug: o uses e_new
  }
  for (int i = 0; i < N_ANG; ++i) {            // RY: [[c,-s],[s,c]]
    float sa, ca; __sincosf(0.5f * ry[i], &sa, &ca);
    float enx = fmaf(ca, ex, -sa * ox), eny = fmaf(ca, ey, -sa * oy);
    float onx = fmaf(sa, enx,  ca * ox), ony = fmaf(sa, eny, ca * oy);
    ex = enx; ey = eny; ox = onx; oy = ony;
  }
  for (int i = 0; i < N_ANG; ++i) {            // RZ: diag(e^-ia/2, e^+ia/2)
    float sa, ca; __sincosf(0.5f * rz[i], &sa, &ca);
    float enx = fmaf(ca, ex,  sa * ey), eny = fmaf(ca, ey, -sa * ex);
    float onx = fmaf(ca, ox, -sa * oy), ony = fmaf(ca, oy,  sa * ox);
    ex = enx; ey = eny; ox = onx; oy = ony;
  }

  // ---- 3. Noise + magnitudes: 8 groups of 4 consecutive elements/thread.
  const uint32_t rowbase = (uint32_t)b * (uint32_t)DIM;
  const bool first_half = (rowbase < half_n);  // uniform per block
  float mag[32];
  float partial = 0.0f;

#pragma unroll
  for (int k = 0; k < 8; ++k) {
#pragma unroll
    for (int j = 0; j < 4; ++j) {
      const int idx = (t << 2) + (k << 10) + j;       // element within row
      const uint32_t m = rowbase + (uint32_t)idx;     // global element id
      const uint32_t x0 = first_half ? m : (m - half_n);
      const uint32_t x1 = first_half ? (m + half_n) : m;
      const U2 dr = threefry2x32(kre0, kre1, x0, x1);
      const U2 di = threefry2x32(kim0, kim1, x0, x1);
      const uint32_t bre = first_half ? dr.a : dr.b;
      const uint32_t bim = first_half ? di.a : di.b;
      // (re + i*im)/sqrt2 * sqrt2*erfinv * 0.01  ==  0.01*erfinv per part
      float re = DECO * erfinv_f(jax_uniform_pm1(bre));
      float im = DECO * erfinv_f(jax_uniform_pm1(bim));
      if (idx == 0) { re += ex; im += ey; }
      else if (idx == 1) { re += ox; im += oy; }
      const float m2 = fmaf(re, re, im * im);
      partial += m2;
      mag[4 * k + j] = __fsqrt_rn(m2);
    }
  }

  // ---- 4. Norm reduction. WMMA folds (L, L+16) lane pairs:
  //   A = ones(16x4), B holds partials in VGPR b0 ->
  //   D[m,n] = p[n] + p[n+16] (uniform in m, and in which k-rows b0 maps to).
  float s;
#if defined(__gfx1250__) && __has_builtin(__builtin_amdgcn_wmma_f32_16x16x4_f32)
  {
    v2f A; A[0] = 1.0f; A[1] = 1.0f;
    v2f B; B[0] = partial; B[1] = 0.0f;
    v8f C = {};
    v8f D = __builtin_amdgcn_wmma_f32_16x16x4_f32(
        /*neg_a=*/false, A, /*neg_b=*/false, B,
        /*c_mod=*/(short)0, C, /*reuse_a=*/false, /*reuse_b=*/false);
    s = D[0];                       // lane L: p[L%16] + p[L%16 + 16]
  }
#else
  s = partial + __shfl_xor(partial, 16, 32);
#endif
  s += __shfl_xor(s, 8, 32);
  s += __shfl_xor(s, 4, 32);
  s += __shfl_xor(s, 2, 32);
  s += __shfl_xor(s, 1, 32);        // full wave32 sum in every lane

  __shared__ float red[8];
  const int wave = t >> 5;
  if ((t & 31) == 0) red[wave] = s;
  __syncthreads();
  float norm2 = red[0] + red[1] + red[2] + red[3] +
                red[4] + red[5] + red[6] + red[7];
  const float inv = __frsqrt_rn(norm2);

  // ---- 5. Coalesced float4 stores: out = mag * rsqrt(norm2).
  float4* orow = reinterpret_cast<float4*>(out + (size_t)b * DIM);
#pragma unroll
  for (int k = 0; k < 8; ++k) {
    float4 v;
    v.x = mag[4 * k + 0] * inv;
    v.y = mag[4 * k + 1] * inv;
    v.z = mag[4 * k + 2] * inv;
    v.w = mag[4 * k + 3] * inv;
    orow[t + (k << 8)] = v;
  }
}

extern "C" void kernel_launch(void* const* d_in, const int* in_sizes, int n_in,
                              void* d_out, int out_size, void* d_ws, size_t ws_size,
                              hipStream_t stream) {
  // Inputs (setup_inputs order): x [B,128] (UNUSED by the reference math),
  // rx_params [13,2], ry_params [13,2], rz_params [13,2]. Output: [B, 8192] f32.
  (void)n_in; (void)d_ws; (void)ws_size; (void)out_size;
  const float* rx = (const float*)d_in[1];
  const float* ry = (const float*)d_in[2];
  const float* rz = (const float*)d_in[3];
  float* out = (float*)d_out;

  const int batch = in_sizes[0] / 128;              // 256
  const uint32_t half_n = (uint32_t)batch * (uint32_t)DIM / 2u;

  hipLaunchKernelGGL(quantum_layer_kernel, dim3(batch), dim3(THREADS), 0, stream,
                     rx, ry, rz, out, half_n);
}